// DependencyGenerator_33938831573598
// MI455X (gfx1250) — compile-verified
//
#include <hip/hip_runtime.h>
#include <stdint.h>

// Four async LDS->global b128 stores per thread per tile, using immediate
// offsets so address math is done once per tile.  Lanes are contiguous
// (16B each -> 512B per wave per instruction); the four offsets step by
// 4096B = one full 256-thread block width, so a block writes a contiguous
// 16KB tile per iteration.
#define ASYNC_FILL_STORE(gaddr, ldsoff, OFS)                                  \
    asm volatile("global_store_async_from_lds_b128 %0, %1, off offset:" #OFS  \
                 :: "v"(gaddr), "v"(ldsoff) : "memory")

// ---------------------------------------------------------------------------
// Kernel 1: fill d_out with 1.0f via gfx1250 async LDS->global stores.
// One 16B vector of ones is staged in LDS (broadcast-read by all lanes);
// each lane issues b128 async stores tracked on ASYNCcnt, drained once with
// s_wait_asynccnt 0.
// ---------------------------------------------------------------------------
__global__ void fill_ones_async_kernel(float* __restrict__ out,
                                       unsigned n4, unsigned ntail) {
    __shared__ __attribute__((aligned(16))) float ones[4];
    if (threadIdx.x == 0) {
        ones[0] = 1.0f; ones[1] = 1.0f; ones[2] = 1.0f; ones[3] = 1.0f;
    }
    __syncthreads();

    // Low 32 bits of the generic address of an LDS object = group-segment
    // byte offset on CDNA5 (aperture lives in the high bits).
    unsigned lds_off = (unsigned)(uintptr_t)(&ones[0]);

    unsigned long long base = (unsigned long long)(uintptr_t)out;

    // 16KB tiles: 256 threads * 4 float4-slots (strided 256 float4 apart).
    unsigned nchunks = (n4 + 1023u) >> 10;          // ceil(n4 / 1024)
    for (unsigned c = blockIdx.x; c < nchunks; c += gridDim.x) {
        unsigned i0 = (c << 10) + threadIdx.x;      // float4 index, lane-contig
        unsigned long long g = base + 16ull * (unsigned long long)i0;
        if (i0 + 768u < n4) {                       // full interior tile
            ASYNC_FILL_STORE(g, lds_off, 0);
            ASYNC_FILL_STORE(g, lds_off, 4096);
            ASYNC_FILL_STORE(g, lds_off, 8192);
            ASYNC_FILL_STORE(g, lds_off, 12288);
        } else {                                    // boundary tile: guard each
            if (i0           < n4) ASYNC_FILL_STORE(g, lds_off, 0);
            if (i0 +  256u   < n4) ASYNC_FILL_STORE(g, lds_off, 4096);
            if (i0 +  512u   < n4) ASYNC_FILL_STORE(g, lds_off, 8192);
            if (i0 +  768u   < n4) ASYNC_FILL_STORE(g, lds_off, 12288);
        }
    }

    // Tail (out_size % 4 floats) — plain scalar stores by one thread.
    if (blockIdx.x == 0 && threadIdx.x == 0) {
        for (unsigned k = 0; k < ntail; ++k)
            out[(size_t)n4 * 4 + k] = 1.0f;
    }

    // Drain ASYNCcnt (S_ENDPGM also implies wait-idle; explicit for clarity).
    asm volatile("s_wait_asynccnt 0x0" ::: "memory");
}

// ---------------------------------------------------------------------------
// Kernel 2: scatter dep_emb[dep_type] into out[b, dep_i, dep_j].
// 65,408 threads; gathers from the 53-entry table (L2-resident).
// ---------------------------------------------------------------------------
__global__ void dep_scatter_kernel(const int* __restrict__ dep_i,
                                   const int* __restrict__ dep_j,
                                   const int* __restrict__ dep_type,
                                   const int* __restrict__ seq_len_p,
                                   const float* __restrict__ dep_emb,
                                   float* __restrict__ out,
                                   int total) {
    int t = blockIdx.x * blockDim.x + threadIdx.x;
    if (t >= total) return;

    int L   = *seq_len_p;        // uniform -> scalar load
    int Lm1 = L - 1;

    int b  = t / Lm1;            // batch index
    int i  = dep_i[t];
    int j  = dep_j[t];
    int ty = dep_type[t];
    float v = dep_emb[ty];       // (53,1) table, flat gather

    size_t off = (size_t)b * (size_t)L * (size_t)L
               + (size_t)i * (size_t)L + (size_t)j;
    out[off] = v;
}

extern "C" void kernel_launch(void* const* d_in, const int* in_sizes, int n_in,
                              void* d_out, int out_size, void* d_ws, size_t ws_size,
                              hipStream_t stream) {
    const int*   dep_i    = (const int*)d_in[0];
    const int*   dep_j    = (const int*)d_in[1];
    const int*   dep_type = (const int*)d_in[2];
    const int*   seq_len  = (const int*)d_in[3];
    const float* dep_emb  = (const float*)d_in[4];
    float*       out      = (float*)d_out;

    // ---- Fill pass: 128 MB of 1.0f, 4 x b128 async stores per thread/tile ----
    unsigned n4    = (unsigned)(out_size >> 2);   // number of float4 chunks
    unsigned ntail = (unsigned)(out_size & 3);
    dim3 fill_block(256);
    dim3 fill_grid(4096);  // 8192 tiles of 16KB -> 2 tiles (8 async stores) per block
    fill_ones_async_kernel<<<fill_grid, fill_block, 0, stream>>>(out, n4, ntail);

    // ---- Scatter pass: one thread per (b, k) dependency entry ----
    int total = in_sizes[0];                      // b * (L-1)
    dim3 sc_block(256);
    dim3 sc_grid((unsigned)((total + 255) / 256));
    dep_scatter_kernel<<<sc_grid, sc_block, 0, stream>>>(
        dep_i, dep_j, dep_type, seq_len, dep_emb, out, total);
}